// SpdRotGr_29987461660843
// MI455X (gfx1250) — compile-verified
//
#include <hip/hip_runtime.h>
#include <math.h>

#if __has_builtin(__builtin_amdgcn_wmma_f32_16x16x4_f32)
#define HAVE_WMMA_F32X4 1
#endif

#if __has_builtin(__builtin_amdgcn_tensor_load_to_lds)
#define HAVE_TDM 1
#endif

typedef float v2f __attribute__((ext_vector_type(2)));
typedef float v8f __attribute__((ext_vector_type(8)));
typedef unsigned int u32x4 __attribute__((ext_vector_type(4)));
typedef int i32x4 __attribute__((ext_vector_type(4)));
typedef int i32x8 __attribute__((ext_vector_type(8)));

#define T_SEQ 24
#define NS_ITERS 12
#define LOG_LEVELS 4

// ---------------------------------------------------------------------------
// Tensor Data Mover: async 1-D DMA of `nelems` f32 from global -> LDS.
// D# built per CDNA5 ISA:
//  group0: count=1 | lds_addr | global_addr[56:0] | type=2
//  group1: data_size=4B, tensor_dim0=tile_dim0=nelems, dim0_stride=nelems,
//          tensor_dim1=1, no multicast / pad / iterate / gather.
// Completion tracked with TENSORcnt (s_wait_tensorcnt, immediate operand).
// ---------------------------------------------------------------------------
#ifdef HAVE_TDM
static __device__ __forceinline__ void tdm_load_1d(float* lds_dst, const float* gsrc,
                                                   int nelems) {
  unsigned long long ga = (unsigned long long)gsrc;
  u32x4 g0;
  g0[0] = 1u;                                            // count=1, user mode
  g0[1] = (unsigned)(unsigned long long)lds_dst;         // lds_addr (bytes)
  g0[2] = (unsigned)(ga & 0xffffffffull);                // global_addr[95:64]
  g0[3] = (unsigned)((ga >> 32) & 0x1ffffffull) | (2u << 30);  // addr[120:96] | type=2
  i32x8 g1 = {};
  g1[0] = 2 << 16;                                       // data_size = 4 bytes
  g1[1] = (nelems & 0xffff) << 16;                       // tensor_dim0[15:0] @ bit48
  g1[2] = ((nelems >> 16) & 0xffff) | (1 << 16);         // tensor_dim0[31:16] | tensor_dim1=1
  g1[3] = (nelems & 0xffff) << 16;                       // tile_dim0 @ bit112
  g1[4] = 0;                                             // tile_dim1/2 = 0 (1-D tile)
  g1[5] = nelems;                                        // tensor_dim0_stride[31:0]
  g1[6] = (nelems & 0xffff) << 16;                       // tensor_dim1_stride[15:0]
  g1[7] = 0;
  i32x4 z4 = {};
#if defined(__clang_major__) && (__clang_major__ >= 23)
  i32x8 z8 = {};
  __builtin_amdgcn_tensor_load_to_lds(g0, g1, z4, z4, z8, 0);
#else
  __builtin_amdgcn_tensor_load_to_lds(g0, g1, z4, z4, 0);
#endif
}
template <int N>
static __device__ __forceinline__ void tdm_wait() {
#if __has_builtin(__builtin_amdgcn_sched_barrier)
  __builtin_amdgcn_sched_barrier(0);
#endif
  __builtin_amdgcn_s_wait_tensorcnt(N);    // SIMM16: must be a constant integer
#if __has_builtin(__builtin_amdgcn_sched_barrier)
  __builtin_amdgcn_sched_barrier(0);
#endif
}
#endif  // HAVE_TDM

// ---------------------------------------------------------------------------
// WMMA f32 16x16x16 matmul built from 4x V_WMMA_F32_16X16X4_F32.
// Tiles live in LDS, row-major. Fragment layouts per CDNA5 ISA 7.12.2 (wave32).
// ---------------------------------------------------------------------------
template <bool BT>
static __device__ __forceinline__ v8f frag_mm(v8f c, const float* __restrict__ A, int lda,
                                              const float* __restrict__ B, int ldb) {
  const int lane = threadIdx.x & 31;
  const int hh = lane >> 4;
  const int mn = lane & 15;
#pragma unroll
  for (int k4 = 0; k4 < 4; ++k4) {
    const int kb = k4 * 4 + hh * 2;
    v2f a, b;
    a.x = A[mn * lda + kb];
    a.y = A[mn * lda + kb + 1];
    if (!BT) {                       // B[k][n]
      b.x = B[kb * ldb + mn];
      b.y = B[(kb + 1) * ldb + mn];
    } else {                         // (B^T)[k][n] = B[n][k]
      b.x = B[mn * ldb + kb];
      b.y = B[mn * ldb + kb + 1];
    }
#ifdef HAVE_WMMA_F32X4
    c = __builtin_amdgcn_wmma_f32_16x16x4_f32(false, a, false, b, (short)0, c, false, false);
#else
#pragma unroll
    for (int r = 0; r < 8; ++r) c[r] += a.x * b.x + a.y * b.y;  // placeholder
#endif
  }
  return c;
}

static __device__ __forceinline__ void store_cfrag(float* D, int ldd, v8f c) {
  const int lane = threadIdx.x & 31;
  const int n = lane & 15;
  const int mb = (lane >> 4) * 8;
#pragma unroll
  for (int r = 0; r < 8; ++r) D[(mb + r) * ldd + n] = c[r];
}

template <bool BT>
static __device__ void mm16t(float* D, const float* A, const float* B) {
  __syncthreads();
  v8f c = {};
  c = frag_mm<BT>(c, A, 16, B, 16);
  __syncthreads();
  store_cfrag(D, 16, c);
  __syncthreads();
}
static __device__ void mm16(float* D, const float* A, const float* B) { mm16t<false>(D, A, B); }
static __device__ void mm16_bt(float* D, const float* A, const float* B) { mm16t<true>(D, A, B); }

// 32x32 matmul as 2x2 blocks of 16x16 WMMA tiles (stride-32 LDS tiles)
template <bool BT>
static __device__ void mm32t(float* D, const float* A, const float* B) {
  __syncthreads();
  v8f acc[2][2];
#pragma unroll
  for (int bi = 0; bi < 2; ++bi)
#pragma unroll
    for (int bj = 0; bj < 2; ++bj) {
      v8f c = {};
#pragma unroll
      for (int bk = 0; bk < 2; ++bk) {
        const float* Ab = A + bi * 16 * 32 + bk * 16;
        const float* Bb = BT ? (B + bj * 16 * 32 + bk * 16)   // (B^T) block (bk,bj)
                             : (B + bk * 16 * 32 + bj * 16);
        c = frag_mm<BT>(c, Ab, 32, Bb, 32);
      }
      acc[bi][bj] = c;
    }
  __syncthreads();
#pragma unroll
  for (int bi = 0; bi < 2; ++bi)
#pragma unroll
    for (int bj = 0; bj < 2; ++bj)
      store_cfrag(D + bi * 16 * 32 + bj * 16, 32, acc[bi][bj]);
  __syncthreads();
}

// ---------------------------------------------------------------------------
// Elementwise tile helpers (one wave owns the tile; strided by lane)
// ---------------------------------------------------------------------------
static __device__ void t16_scale(float* D, const float* S, float sc) {
  __syncthreads();
  for (int i = threadIdx.x; i < 256; i += 32) D[i] = sc * S[i];
  __syncthreads();
}
static __device__ void t16_copy(float* D, const float* S) { t16_scale(D, S, 1.f); }
static __device__ void t16_ident(float* D) {
  __syncthreads();
  for (int i = threadIdx.x; i < 256; i += 32) D[i] = ((i >> 4) == (i & 15)) ? 1.f : 0.f;
  __syncthreads();
}
static __device__ void t32_ident(float* D) {
  __syncthreads();
  for (int i = threadIdx.x; i < 1024; i += 32) D[i] = ((i >> 5) == (i & 31)) ? 1.f : 0.f;
  __syncthreads();
}

static __device__ float wave_sum(float v) {
#pragma unroll
  for (int off = 16; off > 0; off >>= 1) v += __shfl_down(v, off, 32);
  return __shfl(v, 0, 32);
}
static __device__ float frob2(const float* A, int n) {
  __syncthreads();
  float s = 0.f;
  for (int i = threadIdx.x; i < n; i += 32) { float v = A[i]; s += v * v; }
  return wave_sum(s);
}
static __device__ float frob2_diff(const float* A, const float* B, int n) {
  __syncthreads();
  float s = 0.f;
  for (int i = threadIdx.x; i < n; i += 32) { float v = A[i] - B[i]; s += v * v; }
  return wave_sum(s);
}

// D = sym(S) for a 16x16 row-major source (global or LDS staging)
static __device__ void sym_tile(float* D, const float* __restrict__ E) {
  __syncthreads();
#pragma unroll
  for (int q = 0; q < 8; ++q) {
    int i = (threadIdx.x & 31) * 8 + q;
    D[i] = 0.5f * (E[i] + E[(i & 15) * 16 + (i >> 4)]);
  }
  __syncthreads();
}

// E = expm(s*X) ~ I + Y + Y^2/2 + Y^3/6 , Y = s*X (token tangents are ~1e-3)
static __device__ void expm16(float* E, const float* X, float s, float* W1, float* W2) {
  t16_scale(W1, X, s);
  mm16(W2, W1, W1);
  mm16(E, W2, W1);
  __syncthreads();
  for (int i = threadIdx.x; i < 256; i += 32) {
    float id = ((i >> 4) == (i & 15)) ? 1.f : 0.f;
    E[i] = id + W1[i] + 0.5f * W2[i] + (1.f / 6.f) * E[i];
  }
  __syncthreads();
}

// Newton-Schulz: sq -> sqrtm(A), isq -> invsqrtm(A) (A SPD, Frobenius-normalized)
static __device__ void ns16(const float* A, float* Y, float* Z, float* Y2, float* Z2, float* TT,
                            float** sqOut, float** isqOut) {
  float c = sqrtf(frob2(A, 256)) + 1e-30f;
  t16_scale(Y, A, 1.f / c);
  t16_ident(Z);
  float *Yc = Y, *Zc = Z, *Yn = Y2, *Zn = Z2;
#pragma unroll 1
  for (int it = 0; it < NS_ITERS; ++it) {
    mm16(TT, Zc, Yc);
    __syncthreads();
    for (int i = threadIdx.x; i < 256; i += 32) {
      float id = ((i >> 4) == (i & 15)) ? 1.5f : 0.f;
      TT[i] = id - 0.5f * TT[i];
    }
    __syncthreads();
    mm16(Yn, Yc, TT);
    mm16(Zn, TT, Zc);
    float* t = Yc; Yc = Yn; Yn = t;
    t = Zc; Zc = Zn; Zn = t;
  }
  float sc = sqrtf(c);
  t16_scale(Yc, Yc, sc);
  t16_scale(Zc, Zc, 1.f / sc);
  *sqOut = Yc; *isqOut = Zc;
}

// addition_tangents: Y = s0 (s1 (... exp(x_{T-1}) ...) s1) s0, s_t = expm(0.5 x_t)
// Token rows double-buffered into LDS via TDM, overlapped with the WMMA chain.
static __device__ void spd_chain(float* Y, const int* __restrict__ ids,
                                 const float* __restrict__ Emb,
                                 float* X, float* S, float* P, float* W1, float* W2,
                                 float* stage /* 2 x 256 f32 staging slots */) {
#ifdef HAVE_TDM
  tdm_load_1d(stage, Emb + (size_t)ids[T_SEQ - 1] * 256, 256);
  tdm_load_1d(stage + 256, Emb + (size_t)ids[T_SEQ - 2] * 256, 256);
  tdm_wait<1>();                                // slot0 (token T-1) landed
  sym_tile(X, stage);                           // ends with full DS retire
  tdm_load_1d(stage, Emb + (size_t)ids[T_SEQ - 3] * 256, 256);
  expm16(Y, X, 1.0f, W1, W2);
#pragma unroll 1
  for (int t = T_SEQ - 2; t >= 0; --t) {
    const int s = (T_SEQ - 1 - t) & 1;          // slot holding token t
    if (t >= 1) tdm_wait<1>(); else tdm_wait<0>();
    sym_tile(X, stage + 256 * s);
    if (t >= 2) tdm_load_1d(stage + 256 * s, Emb + (size_t)ids[t - 2] * 256, 256);
    expm16(S, X, 0.5f, W1, W2);
    mm16(P, S, Y);
    mm16(Y, P, S);
  }
#else
  (void)stage;
  sym_tile(X, Emb + (size_t)ids[T_SEQ - 1] * 256);
  expm16(Y, X, 1.0f, W1, W2);
#pragma unroll 1
  for (int t = T_SEQ - 2; t >= 0; --t) {
    if (t > 0) __builtin_prefetch(Emb + (size_t)ids[t - 1] * 256, 0, 1);
    sym_tile(X, Emb + (size_t)ids[t] * 256);
    expm16(S, X, 0.5f, W1, W2);
    mm16(P, S, Y);
    mm16(Y, P, S);
  }
#endif
}

// ||logm(M)||_F via inverse scaling (LOG_LEVELS NS square roots) + Mercator series
static __device__ float spd_log_dist(float* M, float* w1, float* w2, float* w3, float* w4, float* w5) {
#pragma unroll 1
  for (int l = 0; l < LOG_LEVELS; ++l) {
    float *sq, *isq;
    ns16(M, w1, w2, w3, w4, w5, &sq, &isq);
    t16_copy(M, sq);
  }
  __syncthreads();
  for (int i = threadIdx.x; i < 256; i += 32)
    M[i] -= ((i >> 4) == (i & 15)) ? 1.f : 0.f;        // E = M^(1/2^L) - I
  __syncthreads();
  mm16(w1, M, M);     // E^2
  mm16(w2, w1, M);    // E^3
  mm16(w3, w1, w1);   // E^4
  __syncthreads();
  for (int i = threadIdx.x; i < 256; i += 32)
    w4[i] = M[i] - 0.5f * w1[i] + (1.f / 3.f) * w2[i] - 0.25f * w3[i];
  float d2 = frob2(w4, 256);
  return sqrtf(fmaxf(d2, 0.f)) * (float)(1 << LOG_LEVELS);
}

// 20x20 skew tangent embedded in a 32x32 tile (pad = 0; identity added in cayley)
static __device__ void load_skew32(float* A, const float* __restrict__ Eg,
                                   const float* __restrict__ tf) {
  __syncthreads();
  for (int i = threadIdx.x; i < 1024; i += 32) A[i] = 0.f;
  __syncthreads();
  for (int i = threadIdx.x; i < 100; i += 32) {
    int r = i / 10, c = i % 10;
    float x = Eg[i];
    if (tf) x *= tf[i];
    A[r * 32 + 10 + c] = x;
    A[(10 + c) * 32 + r] = -x;
  }
  __syncthreads();
}

// z = prod_t cayley(A_t);  cayley(A) = (I-A)(I+A)^-1 = I - 2A + 2A^2 - 2A^3 + O(A^4)
static __device__ float* gr_chain(float* z0, float* z1, float* Yg, float* Ag, float* A2,
                                  const int* __restrict__ ids, const float* __restrict__ Eg,
                                  const float* __restrict__ tf,
                                  float* stage /* 2 x 128 f32 staging slots */) {
  t32_ident(z0);
  float *zc = z0, *zx = z1;
#ifdef HAVE_TDM
  tdm_load_1d(stage, Eg + (size_t)ids[0] * 100, 100);
  tdm_load_1d(stage + 128, Eg + (size_t)ids[1] * 100, 100);
#endif
#pragma unroll 1
  for (int t = 0; t < T_SEQ; ++t) {
#ifdef HAVE_TDM
    const int s = t & 1;
    if (t < T_SEQ - 1) tdm_wait<1>(); else tdm_wait<0>();
    load_skew32(Ag, stage + 128 * s, tf);       // ends with full DS retire
    if (t + 2 < T_SEQ) tdm_load_1d(stage + 128 * s, Eg + (size_t)ids[t + 2] * 100, 100);
#else
    if (t + 1 < T_SEQ) __builtin_prefetch(Eg + (size_t)ids[t + 1] * 100, 0, 1);
    load_skew32(Ag, Eg + (size_t)ids[t] * 100, tf);
#endif
    mm32t<false>(A2, Ag, Ag);        // A^2
    mm32t<false>(Yg, A2, Ag);        // A^3
    __syncthreads();
    for (int i = threadIdx.x; i < 1024; i += 32) {
      float id = ((i >> 5) == (i & 31)) ? 1.f : 0.f;
      Yg[i] = id - 2.f * Ag[i] + 2.f * A2[i] - 2.f * Yg[i];
    }
    __syncthreads();
    mm32t<false>(zx, zc, Yg);
    float* tp = zc; zc = zx; zx = tp;
  }
  return zc;
}

// z @ P (P = diag(1x10,0)) == zero columns >= 10
static __device__ void mask_cols10(float* D, const float* Z) {
  __syncthreads();
  for (int i = threadIdx.x; i < 1024; i += 32) D[i] = ((i & 31) < 10) ? Z[i] : 0.f;
  __syncthreads();
}

// ---------------------------------------------------------------------------
// Prelude: batch-invariant constants -> ws  [rel(256) | expB(256) | ybias(1024)]
// ---------------------------------------------------------------------------
__global__ __launch_bounds__(32) void geo_prelude(
    const float* __restrict__ refp, const float* __restrict__ biasS,
    const float* __restrict__ biasG, float* __restrict__ wsf) {
  __shared__ float R[256];
  __shared__ float W0[256], W1[256], W2[256], W3[256];
  __shared__ float A20[400], M20[400], I20[400];
  const int lane = threadIdx.x;

  // --- rel = product of 120 Givens rotations (right-multiplies touch 2 cols) ---
  for (int i = lane; i < 256; i += 32) R[i] = ((i >> 4) == (i & 15)) ? 1.f : 0.f;
  __syncthreads();
  int gi = 0, gj = 1;
#pragma unroll 1
  for (int k = 0; k < 120; ++k) {
    float ang = refp[k];
    float cs = cosf(ang), sn = sinf(ang);
    if (lane < 16) {
      float a = R[lane * 16 + gi];
      float c = R[lane * 16 + gj];
      R[lane * 16 + gi] = cs * a + sn * c;
      R[lane * 16 + gj] = -sn * a + cs * c;
    }
    __syncthreads();
    if (++gj == 16) { ++gi; gj = gi + 1; }
  }
  for (int i = lane; i < 256; i += 32) wsf[i] = R[i];
  __syncthreads();

  // --- expB = expm(sym(biasS)) : scale 2^-6, Taylor(8), square 6x (WMMA) ---
  for (int i = lane; i < 256; i += 32) {
    int r = i >> 4, c = i & 15;
    W0[i] = 0.5f * (biasS[i] + biasS[c * 16 + r]) * (1.f / 64.f);
  }
  __syncthreads();
  for (int i = lane; i < 256; i += 32) {
    float id = ((i >> 4) == (i & 15)) ? 1.f : 0.f;
    W1[i] = W0[i];
    W2[i] = id + W0[i];
  }
  __syncthreads();
#pragma unroll 1
  for (int k = 2; k <= 8; ++k) {
    mm16(W3, W1, W0);
    __syncthreads();
    for (int i = lane; i < 256; i += 32) {
      W1[i] = W3[i] * (1.f / (float)k);
      W2[i] += W1[i];
    }
    __syncthreads();
  }
#pragma unroll 1
  for (int k = 0; k < 6; ++k) {
    mm16(W3, W2, W2);
    t16_copy(W2, W3);
  }
  for (int i = lane; i < 256; i += 32) wsf[256 + i] = W2[i];
  __syncthreads();

  // --- ybias = cayley(skew(biasG)) exact: (I-A)(I+A)^-1, Gauss-Jordan on 20x20 ---
  if (lane == 0) {
    for (int i = 0; i < 400; ++i) A20[i] = 0.f;
    for (int r = 0; r < 10; ++r)
      for (int c = 0; c < 10; ++c) {
        float x = biasG[r * 10 + c];
        A20[r * 20 + 10 + c] = x;
        A20[(10 + c) * 20 + r] = -x;
      }
    for (int i = 0; i < 400; ++i) {
      float id = ((i / 20) == (i % 20)) ? 1.f : 0.f;
      M20[i] = id + A20[i];
      I20[i] = id;
    }
    for (int k = 0; k < 20; ++k) {
      float ip = 1.f / M20[k * 20 + k];
      for (int j = 0; j < 20; ++j) { M20[k * 20 + j] *= ip; I20[k * 20 + j] *= ip; }
      for (int r = 0; r < 20; ++r) {
        if (r == k) continue;
        float f = M20[r * 20 + k];
        for (int j = 0; j < 20; ++j) {
          M20[r * 20 + j] -= f * M20[k * 20 + j];
          I20[r * 20 + j] -= f * I20[k * 20 + j];
        }
      }
    }
  }
  __syncthreads();
  for (int i = lane; i < 1024; i += 32) {   // Yb = (I-A) @ inv(I+A), pad diag = I
    int r = i >> 5, c = i & 31;
    float v;
    if (r < 20 && c < 20) {
      float s = 0.f;
      for (int k = 0; k < 20; ++k) {
        float ia = ((r == k) ? 1.f : 0.f) - A20[r * 20 + k];
        s += ia * I20[k * 20 + c];
      }
      v = s;
    } else {
      v = (r == c) ? 1.f : 0.f;
    }
    wsf[512 + i] = v;
  }
}

// ---------------------------------------------------------------------------
// Main: one wave32 per batch element. Cayley(skew) is orthogonal, so all
// linalg.inv in the Grassmann branch become transposed-B WMMA reads.
// TDM double-buffers token tiles into idle LDS while WMMA chains run.
// ---------------------------------------------------------------------------
__global__ __launch_bounds__(32) void geo_main(
    const int* __restrict__ qids, const int* __restrict__ aids,
    const float* __restrict__ Eq, const float* __restrict__ Ea,
    const float* __restrict__ Eqg, const float* __restrict__ Eag,
    const float* __restrict__ tf, const float* __restrict__ wfp,
    const float* __restrict__ wbp, const float* __restrict__ wsf,
    float* __restrict__ out) {
  __shared__ float t16[7][256];
  __shared__ float t32[6][1024];
  const int b = blockIdx.x;
  const float* rel = wsf;
  const float* expB = wsf + 256;
  const float* ybias = wsf + 512;

  // ===== SPD branch ===== (t32[0] doubles as the 2x256 TDM staging area)
  spd_chain(t16[0], qids + b * T_SEQ, Eq, t16[1], t16[2], t16[3], t16[4], t16[5], t32[0]);
  mm16(t16[1], rel, t16[0]);
  mm16_bt(t16[0], t16[1], rel);                       // rel @ Y @ rel^T
  float *sq, *isq;
  ns16(t16[0], t16[1], t16[2], t16[3], t16[4], t16[5], &sq, &isq);
  mm16(t16[6], sq, expB);
  mm16(t16[0], t16[6], sq);                           // q_spd = sqrt(Y) expB sqrt(Y)
  ns16(t16[0], t16[1], t16[2], t16[3], t16[4], t16[5], &sq, &isq);
  t16_copy(t16[6], isq);                              // park invsqrt(q_spd)
  spd_chain(t16[0], aids + b * T_SEQ, Ea, t16[1], t16[2], t16[3], t16[4], t16[5], t32[0]);
  mm16(t16[1], t16[6], t16[0]);
  mm16(t16[0], t16[1], t16[6]);                       // M = isq a_spd isq
  float dspd = spd_log_dist(t16[0], t16[1], t16[2], t16[3], t16[4], t16[5]);

  // ===== Grassmann branch ===== (t16[0] doubles as the 2x128 TDM staging area)
  float* za = gr_chain(t32[0], t32[1], t32[2], t32[3], t32[4],
                       aids + b * T_SEQ, Eag, nullptr, t16[0]);
  mask_cols10(t32[2], za);
  mm32t<true>(t32[5], t32[2], t32[2]);                // a_gr = (zP)(zP)^T
  float* zq = gr_chain(t32[0], t32[1], t32[2], t32[3], t32[4],
                       qids + b * T_SEQ, Eqg, tf, t16[0]);
  mask_cols10(t32[2], zq);
  mm32t<true>(t32[4], t32[2], t32[2]);                // G = (zP)(zP)^T
  mm32t<false>(t32[3], ybias, t32[4]);
  mm32t<true>(t32[4], t32[3], ybias);                 // q_gr = Yb G Yb^T
  float dgr = sqrtf(fmaxf(frob2_diff(t32[4], t32[5], 1024), 0.f));

  if (threadIdx.x == 0) out[b] = -wfp[0] * (dspd + dgr) + wbp[0];
}

// ---------------------------------------------------------------------------
extern "C" void kernel_launch(void* const* d_in, const int* in_sizes, int n_in,
                              void* d_out, int out_size, void* d_ws, size_t ws_size,
                              hipStream_t stream) {
  (void)in_sizes; (void)n_in; (void)ws_size;
  const int* qids = (const int*)d_in[0];
  const int* aids = (const int*)d_in[1];
  const float* Eq = (const float*)d_in[2];
  const float* Ea = (const float*)d_in[3];
  const float* refp = (const float*)d_in[4];
  const float* biasS = (const float*)d_in[5];
  const float* Eqg = (const float*)d_in[6];
  const float* Eag = (const float*)d_in[7];
  const float* tfg = (const float*)d_in[8];
  const float* biasG = (const float*)d_in[9];
  const float* wf = (const float*)d_in[10];
  const float* wb = (const float*)d_in[11];
  float* out = (float*)d_out;
  float* wsf = (float*)d_ws;

  geo_prelude<<<1, 32, 0, stream>>>(refp, biasS, biasG, wsf);
  geo_main<<<out_size, 32, 0, stream>>>(qids, aids, Eq, Ea, Eqg, Eag, tfg, wf, wb, wsf, out);
}